// LiteTensorMapperBlock_1992864825520
// MI455X (gfx1250) — compile-verified
//
#include <hip/hip_runtime.h>
#include <math.h>

typedef __attribute__((ext_vector_type(2))) float v2f;
typedef __attribute__((ext_vector_type(8))) float v8f;
typedef __attribute__((ext_vector_type(4))) unsigned v4u;
typedef __attribute__((ext_vector_type(8))) int v8i;
typedef __attribute__((ext_vector_type(4))) int v4i;

#define WMMA_F32(A, B, C) \
  __builtin_amdgcn_wmma_f32_16x16x4_f32(false, (A), false, (B), (short)0, (C), false, false)

// Sizes for this problem instance.
#define BB 16
#define SS 2048
#define DD 256
#define NTOK (BB * SS)            // 32768 rows
#define TOTAL (BB * SS * DD)      // 8388608 elements
#define WPAD 260                  // padded LDS row stride (bank-conflict free; matches TDM pad)
#define LRSCALE (0.01f * (2.0f / (float)(BB * DD)))  // 0.01 * 2/(B*D)

// ---------------------------------------------------------------------------
// TDM helper: async-load one 16x256 f32 tile (row-major, stride 256) from
// global memory into LDS with 4-DWORD padding after every 256 DWORDs, i.e.
// LDS row stride = 260 floats. Tracked by TENSORcnt.
// D# packing per CDNA5 ISA ch.10.8 (group0/group1); groups 2/3 zero (2D tile).
// ---------------------------------------------------------------------------
__device__ inline unsigned lds_offset_of(const void* p) {
  // generic -> LDS address space; numeric value of an AS(3) pointer is the
  // byte offset within the workgroup's LDS allocation.
  return (unsigned)(unsigned long long)(__attribute__((address_space(3))) const void*)p;
}

__device__ inline void tdm_load_tile16x256(const float* gptr, unsigned lds_off) {
  unsigned long long ga = (unsigned long long)gptr;
  v4u g0;
  g0.x = 1u;                                            // count=1 (valid descriptor)
  g0.y = lds_off;                                       // lds_addr [63:32]
  g0.z = (unsigned)ga;                                  // global_addr[31:0]
  g0.w = (unsigned)((ga >> 32) & 0x01FFFFFFu)           // global_addr[56:32]
         | 0x80000000u;                                 // type=2 ("image") [127:126]
  v8i g1;
  g1[0] = (2 << 16)      // data_size = 4 bytes
        | (1 << 20)      // pad_enable
        | (7 << 22)      // pad_interval: pad after every 256 DWORDs (one row)
        | (3 << 25);     // pad_amount: 4 DWORDs  -> LDS row stride 260 floats
  g1[1] = (int)(256u << 16);   // tensor_dim0 = 256 (bits 79:48, low half)
  g1[2] = (int)(16u << 16);    // tensor_dim0 hi = 0 | tensor_dim1 = 16
  g1[3] = (int)(256u << 16);   // tensor_dim1 hi = 0 | tile_dim0 = 256
  g1[4] = 16;                  // tile_dim1 = 16, tile_dim2 = 0 (2D)
  g1[5] = 256;                 // tensor_dim0_stride = 256 (bits 207:160, low)
  g1[6] = 0;                   // stride0 hi / tensor_dim1_stride lo
  g1[7] = 0;
  v4i z4 = {0, 0, 0, 0};
  v8i z8 = {0, 0, 0, 0, 0, 0, 0, 0};
  // clang-23 / therock arity: (g0, g1, g2, g3, <ext group>, cpol)
  __builtin_amdgcn_tensor_load_to_lds(g0, g1, z4, z4, z8, 0);
}

// ---------------------------------------------------------------------------
// Kernel A: elementwise prep.
//   ws_m[t][b][d]     = tanh(alpha1[d] * x[b][t][d])        (TTT "state", step-major)
//   ws_train[t][b][d] = m + noise[b][t][d]                  (TTT "train", step-major)
//   ws_v[b][t][d]     = tanh(alpha2[d] * x[b][t][d])        (FF GEMM A matrix, row-major)
// ---------------------------------------------------------------------------
__global__ __launch_bounds__(256) void prep_kernel(
    const float* __restrict__ x, const float* __restrict__ noise,
    const float* __restrict__ alpha1, const float* __restrict__ alpha2,
    float* __restrict__ ws_m, float* __restrict__ ws_train, float* __restrict__ ws_v) {
  for (int i = blockIdx.x * blockDim.x + threadIdx.x; i < TOTAL;
       i += gridDim.x * blockDim.x) {
    int d = i & (DD - 1);
    int s = (i >> 8) & (SS - 1);
    int b = i >> 19;                       // i / (SS*DD)
    float xv = x[i];
    float m  = tanhf(alpha1[d] * xv);
    float tr = m + noise[i];
    float vv = tanhf(alpha2[d] * xv);
    int mt = ((s * BB + b) << 8) + d;      // [t][b][d]
    ws_m[mt]     = m;
    ws_train[mt] = tr;
    ws_v[i]      = vv;                     // same layout as x: [b][s][d]
  }
}

// ---------------------------------------------------------------------------
// Kernel B: FF path. g = u .* (v @ W_proj^T), written to out.
// One wave per 16x16 output tile, K=256 via 64x V_WMMA_F32_16X16X4_F32
// (two interleaved accumulators for ILP). Exact-erf GELU epilogue.
// ---------------------------------------------------------------------------
__global__ __launch_bounds__(128) void ff_gemm_kernel(
    const float* __restrict__ v, const float* __restrict__ wproj,
    const float* __restrict__ ff_alpha, const float* __restrict__ ff_beta,
    const float* __restrict__ ff_gamma, const float* __restrict__ ff_eta,
    float* __restrict__ out) {
  const int wave = threadIdx.x >> 5;
  const int lane = threadIdx.x & 31;
  const int nl = lane & 15;
  const int khalf = (lane < 16) ? 0 : 2;

  const int tile = blockIdx.x * 4 + wave;        // 32768 tiles total
  const int tn = tile & 15;                      // 16 col tiles
  const int tm = tile >> 4;                      // 2048 row tiles
  const int mrow = tm * 16 + nl;                 // A: M = lane%16
  const int ncol = tn * 16 + nl;                 // B: N = lane%16

  const float* arow = v + mrow * DD;
  const float* brow = wproj + ncol * DD;         // B(k,n) = W_proj[n][k]

  v8f acc0 = {};
  v8f acc1 = {};
  for (int g = 0; g < 64; g += 2) {
    v2f a0 = *(const v2f*)(arow + 4 * g + khalf);
    v2f b0 = *(const v2f*)(brow + 4 * g + khalf);
    v2f a1 = *(const v2f*)(arow + 4 * (g + 1) + khalf);
    v2f b1 = *(const v2f*)(brow + 4 * (g + 1) + khalf);
    acc0 = WMMA_F32(a0, b0, acc0);
    acc1 = WMMA_F32(a1, b1, acc1);
  }

  const float fa = ff_alpha[ncol];
  const float fb = ff_beta[ncol];
  const float gm = ff_gamma[0];
  const float et = ff_eta[0];
  const int bo = (lane >= 16) ? 8 : 0;
#pragma unroll
  for (int r = 0; r < 8; ++r) {
    int row = tm * 16 + r + bo;                  // C/D: M = r (+8 for high half)
    float vv = v[row * DD + ncol];
    float ga = fa * vv;
    float gelu = 0.5f * ga * (1.0f + erff(ga * 0.70710678118654752f));
    float u = gm * gelu + et * sinf(fb * vv);
    out[row * DD + ncol] = u * (acc0[r] + acc1[r]);
  }
}

// ---------------------------------------------------------------------------
// Kernel C: TTT scan. 16 independent workgroups, each owns 16 rows of W
// (columns jb*16..jb*16+15 of every output) in LDS, runs 2048 sequential
// SGD steps. 4 waves split K (diff/readout) and N (grad/update).
// Per-step train/state tiles are double-buffered into LDS by the Tensor
// Data Mover (prefetch t+1 while computing t) -> step body is pure LDS+WMMA.
// ---------------------------------------------------------------------------
__global__ __launch_bounds__(128, 1) void ttt_kernel(
    const float* __restrict__ ws_m, const float* __restrict__ ws_train,
    const float* __restrict__ w_ttt, float* __restrict__ out) {
  __shared__ float W_s[16 * WPAD];       // W_s[jl][k], padded stride
  __shared__ float trn_s[2][16 * WPAD];  // TDM double buffer: train tile
  __shared__ float st_s[2][16 * WPAD];   // TDM double buffer: state tile
  __shared__ float part[4 * 256];        // per-wave partial 16x16 tiles
  __shared__ float diff_s[256];          // reduced diff [b][jl]

  const int tid = threadIdx.x;
  const int wave = tid >> 5;
  const int lane = tid & 31;
  const int nl = lane & 15;
  const int khalf = (lane < 16) ? 0 : 2;
  const int bo = (lane >= 16) ? 8 : 0;
  const int jb = blockIdx.x;             // which 16-row slice of W we own

  const unsigned trn_off[2] = {lds_offset_of(&trn_s[0][0]), lds_offset_of(&trn_s[1][0])};
  const unsigned st_off[2]  = {lds_offset_of(&st_s[0][0]),  lds_offset_of(&st_s[1][0])};

  // Kick off TDM for step 0 while we load W.
  if (wave == 0) {
    tdm_load_tile16x256(ws_train, trn_off[0]);
    tdm_load_tile16x256(ws_m, st_off[0]);
  }

  // Load our W slice: W_s[jl][k] = W_ttt[jb*16+jl][k]
  for (int i = tid; i < 16 * DD; i += 128) {
    int jl = i >> 8, k = i & 255;
    W_s[jl * WPAD + k] = w_ttt[(jb * 16 + jl) * DD + k];
  }
  if (wave == 0) __builtin_amdgcn_s_wait_tensorcnt(0);
  __syncthreads();

  for (int t = 0; t < SS; ++t) {
    const int cur = t & 1, nxt = cur ^ 1;
    const float* trn = &trn_s[cur][0];   // [b][k] with stride WPAD
    const float* st  = &st_s[cur][0];

    // Prefetch step t+1 into the other buffer (overlaps with this step).
    if (wave == 0 && t + 1 < SS) {
      tdm_load_tile16x256(ws_train + (size_t)(t + 1) * BB * DD, trn_off[nxt]);
      tdm_load_tile16x256(ws_m + (size_t)(t + 1) * BB * DD, st_off[nxt]);
    }

    // ---- Phase 1: partial diff = train @ W^T over K in [wave*64, wave*64+64)
    v8f acc = {};
    for (int g = 0; g < 16; ++g) {
      int k0 = wave * 64 + 4 * g + khalf;
      v2f a = *(const v2f*)(trn + nl * WPAD + k0);      // A(m=b, k)
      v2f b = *(const v2f*)(&W_s[nl * WPAD + k0]);      // B(k, n=jl) = W[jl][k]
      acc = WMMA_F32(a, b, acc);
    }
#pragma unroll
    for (int r = 0; r < 8; ++r)
      part[wave * 256 + (r + bo) * 16 + nl] = acc[r];
    __syncthreads();

    // ---- Phase 2a: reduce partials, subtract state -> diff_s[b*16+jl]
    {
      int e = tid * 2;
#pragma unroll
      for (int q = 0; q < 2; ++q, ++e) {
        int b = e >> 4, jl = e & 15;
        diff_s[e] = part[e] + part[256 + e] + part[512 + e] + part[768 + e]
                    - st[b * WPAD + jb * 16 + jl];
      }
    }
    __syncthreads();

    // ---- Phase 2b: grad = diff^T @ train over N in [wave*64,..); update W
    for (int c = wave * 4; c < wave * 4 + 4; ++c) {
      v8f gacc = {};
#pragma unroll
      for (int gp = 0; gp < 4; ++gp) {
        int b0 = 4 * gp + khalf;
        v2f a;                                          // A(m=jl, k=b) = diff[b][jl]
        a.x = diff_s[b0 * 16 + nl];
        a.y = diff_s[(b0 + 1) * 16 + nl];
        v2f b;                                          // B(k=b, n=kcol) = train[b][kcol]
        int kcol = c * 16 + nl;
        b.x = trn[b0 * WPAD + kcol];
        b.y = trn[(b0 + 1) * WPAD + kcol];
        gacc = WMMA_F32(a, b, gacc);
      }
#pragma unroll
      for (int r = 0; r < 8; ++r) {
        int jl = r + bo, k = c * 16 + nl;
        W_s[jl * WPAD + k] -= LRSCALE * gacc[r];
      }
    }
    __syncthreads();

    // ---- Phase 3: partial readout = state @ Wn^T over our K range
    v8f racc = {};
    for (int g = 0; g < 16; ++g) {
      int k0 = wave * 64 + 4 * g + khalf;
      v2f a = *(const v2f*)(st + nl * WPAD + k0);
      v2f b = *(const v2f*)(&W_s[nl * WPAD + k0]);
      racc = WMMA_F32(a, b, racc);
    }
#pragma unroll
    for (int r = 0; r < 8; ++r)
      part[wave * 256 + (r + bo) * 16 + nl] = racc[r];
    __syncthreads();

    // ---- Reduce and accumulate into out (mem += on top of kernel B's g)
    {
      int e = tid * 2;
      int b = e >> 4, jl = e & 15;
      float s0 = part[e] + part[256 + e] + part[512 + e] + part[768 + e];
      float s1 = part[e + 1] + part[256 + e + 1] + part[512 + e + 1] + part[768 + e + 1];
      float* o = out + ((b * SS + t) * DD + jb * 16 + jl);
      float2 curv = *(float2*)o;
      curv.x += s0;
      curv.y += s1;
      *(float2*)o = curv;
    }

    // Wait for the t+1 TDM transfers, then publish to all waves via barrier.
    if (wave == 0) __builtin_amdgcn_s_wait_tensorcnt(0);
    __syncthreads();
  }
}

// ---------------------------------------------------------------------------
// Launch. Inputs (setup_inputs order):
//  0:x 1:noise 2:alpha1 3:alpha2 4:W_ttt 5:W_proj 6:ff_alpha 7:ff_beta
//  8:ff_gamma 9:ff_eta
// Workspace needs 3 * 32 MB = 96 MB (ws_m, ws_train, ws_v as f32).
// ---------------------------------------------------------------------------
extern "C" void kernel_launch(void* const* d_in, const int* in_sizes, int n_in,
                              void* d_out, int out_size, void* d_ws, size_t ws_size,
                              hipStream_t stream) {
  const float* x      = (const float*)d_in[0];
  const float* noise  = (const float*)d_in[1];
  const float* alpha1 = (const float*)d_in[2];
  const float* alpha2 = (const float*)d_in[3];
  const float* w_ttt  = (const float*)d_in[4];
  const float* w_proj = (const float*)d_in[5];
  const float* ff_a   = (const float*)d_in[6];
  const float* ff_b   = (const float*)d_in[7];
  const float* ff_g   = (const float*)d_in[8];
  const float* ff_e   = (const float*)d_in[9];
  float* out = (float*)d_out;

  float* ws       = (float*)d_ws;
  float* ws_m     = ws;                     // [S][B][D]
  float* ws_train = ws + (size_t)TOTAL;     // [S][B][D]
  float* ws_v     = ws + (size_t)TOTAL * 2; // [B][S][D]

  // A: elementwise prep
  prep_kernel<<<8192, 256, 0, stream>>>(x, noise, alpha1, alpha2, ws_m, ws_train, ws_v);

  // B: FF GEMM + epilogue -> out = g   (32768 tiles, 4 waves per block)
  ff_gemm_kernel<<<8192, 128, 0, stream>>>(ws_v, w_proj, ff_a, ff_b, ff_g, ff_e, out);

  // C: 16 independent TTT column-block scans -> out += mem
  ttt_kernel<<<16, 128, 0, stream>>>(ws_m, ws_train, w_ttt, out);
}